// LossLayer_62921270886657
// MI455X (gfx1250) — compile-verified
//
#include <hip/hip_runtime.h>
#include <hip/hip_bf16.h>
#include <stdint.h>

namespace {
constexpr int   kB    = 16;
constexpr int   kN    = 32;
constexpr int   kA    = 102400;
constexpr int   kBlk  = 256;
constexpr int   kGrid = kA / kBlk;   // 400
constexpr int   kL    = 196;
constexpr float kWingC = 3.0f - 3.0f * 0.91629073187415506f; // 3 - 3*ln(2.5)
}

typedef float v2f __attribute__((ext_vector_type(2)));
typedef float v8f __attribute__((ext_vector_type(8)));

// ---- CDNA5 matrix-unit wave reduction -------------------------------------
// Each lane contributes x into the 16x4 f32 A matrix (lanes 0-15 = rows, K0;
// lanes 16-31 = same rows, K2). B = ones(4x16) so D[m][n] = rowsum(A[m]).
// Sum of a lane's 8 D VGPRs gives sum over M=0..7 (lanes<16) or M=8..15
// (lanes>=16); one shfl_xor(16) completes the exact-f32 32-lane total.
__device__ __forceinline__ float wave_reduce32(float x) {
  v2f a; a[0] = x;    a[1] = 0.0f;
  v2f o; o[0] = 1.0f; o[1] = 1.0f;
  v8f c = {};
  c = __builtin_amdgcn_wmma_f32_16x16x4_f32(false, a, false, o,
                                            (short)0, c, false, false);
  float s = c[0] + c[1] + c[2] + c[3] + c[4] + c[5] + c[6] + c[7];
  s += __shfl_xor(s, 16, 32);
  return s; // total across all 32 lanes, in every lane
}

// Deterministic 256-thread block reduction (8 waves). Valid in thread 0.
__device__ __forceinline__ float block_reduce256(float v, float* sm) {
  __syncthreads();
  float s = wave_reduce32(v);
  if ((threadIdx.x & 31) == 0) sm[threadIdx.x >> 5] = s;
  __syncthreads();
  float tot = 0.0f;
  if (threadIdx.x == 0) {
    for (int i = 0; i < 8; ++i) tot += sm[i];
  }
  return tot;
}

__device__ __forceinline__ unsigned sortKey(float v) {
  unsigned u = __float_as_uint(v);
  return (u & 0x80000000u) ? ~u : (u | 0x80000000u);
}
__device__ __forceinline__ float invKey(unsigned k) {
  unsigned u = (k & 0x80000000u) ? (k ^ 0x80000000u) : ~k;
  return __uint_as_float(u);
}

// ---------------------------------------------------------------------------
__global__ void kinit(unsigned* p, int n) {
  for (int i = blockIdx.x * blockDim.x + threadIdx.x; i < n;
       i += gridDim.x * blockDim.x)
    p[i] = 0u;
}

// Phase 1: IoU vs 32 annotations, pos/neg flags + argmax, counts.
__global__ void kiou(const float* __restrict__ anchors,
                     const float* __restrict__ ann,
                     uint8_t* __restrict__ flags,
                     unsigned* __restrict__ nposA,
                     unsigned* __restrict__ nnegA) {
  const int b = blockIdx.y;
  const int a = blockIdx.x * kBlk + threadIdx.x;
  __shared__ float bx1[kN], by1[kN], bx2[kN], by2[kN], barea[kN];
  __shared__ int   bval[kN];
  __shared__ unsigned sNp, sNn;
  if (threadIdx.x == 0) { sNp = 0u; sNn = 0u; }
  if (threadIdx.x < kN) {
    const float* p = &ann[((size_t)b * kN + threadIdx.x) * 200];
    float x1 = p[0], y1 = p[1], x2 = p[2], y2 = p[3];
    bx1[threadIdx.x] = x1; by1[threadIdx.x] = y1;
    bx2[threadIdx.x] = x2; by2[threadIdx.x] = y2;
    barea[threadIdx.x] = (x2 - x1) * (y2 - y1);
    bval[threadIdx.x] = (x1 > 0.0f) ? 1 : 0;
  }
  __syncthreads();

  const float4 ab = reinterpret_cast<const float4*>(anchors)[a];
  const float aarea = (ab.z - ab.x) * (ab.w - ab.y);
  float m = -3.0e38f;
  int   arg = 0;
#pragma unroll 8
  for (int j = 0; j < kN; ++j) {
    float iw = fminf(ab.z, bx2[j]) - fmaxf(ab.x, bx1[j]);
    float ih = fminf(ab.w, by2[j]) - fmaxf(ab.y, by1[j]);
    iw = fmaxf(iw, 0.0f); ih = fmaxf(ih, 0.0f);
    float inter = iw * ih;
    float ua    = fmaxf(aarea + barea[j] - inter, 1e-8f);
    float iou   = inter / ua;
    float v     = bval[j] ? iou : -1.0f;
    if (v > m) { m = v; arg = j; }  // first max wins (strict >)
  }
  const bool neg = m < 0.4f;
  const bool pos = m >= 0.7f;
  flags[(size_t)b * kA + a] =
      (uint8_t)((neg ? 1u : 0u) | (pos ? 2u : 0u) | ((unsigned)arg << 2));
  if (pos) atomicAdd(&sNp, 1u);
  if (neg) atomicAdd(&sNn, 1u);
  __syncthreads();
  if (threadIdx.x == 0) {
    if (sNp) atomicAdd(&nposA[b], sNp);
    if (sNn) atomicAdd(&nnegA[b], sNn);
  }
}

// Radix-select histogram pass (count-only, integer atomics => deterministic).
__global__ void khist(const float* __restrict__ cls,
                      const uint8_t* __restrict__ flags,
                      unsigned* __restrict__ histCnt,
                      const unsigned* __restrict__ keepA,
                      const unsigned* __restrict__ prefixA, int pass) {
  const int b = blockIdx.y;
  __shared__ unsigned lh[256];
  __shared__ unsigned sKeep, sPrefix;
  if (threadIdx.x == 0) { sKeep = keepA[b]; sPrefix = prefixA[b]; }
  lh[threadIdx.x] = 0u;
  __syncthreads();
  if (pass == 0 || sKeep > 0u) {
    const int a = blockIdx.x * kBlk + threadIdx.x;
    const uint8_t f = flags[(size_t)b * kA + a];
    if (f & 1u) {
      float v = -cls[((size_t)b * kA + a) * 2 + 1];
      unsigned key = sortKey(v);
      bool match = (pass == 0) || ((key >> (32 - 8 * pass)) == sPrefix);
      if (match) atomicAdd(&lh[(key >> (24 - 8 * pass)) & 255u], 1u);
    }
  }
  __syncthreads();
  unsigned c = lh[threadIdx.x];
  if (c) atomicAdd(&histCnt[b * 256 + threadIdx.x], c);
}

// Pick the bin holding the k-th largest; refine prefix; zero histogram.
__global__ void kselect(unsigned* __restrict__ histCnt,
                        const unsigned* __restrict__ nposA,
                        const unsigned* __restrict__ nnegA,
                        unsigned* __restrict__ keepA,
                        unsigned* __restrict__ cntAboveA,
                        unsigned* __restrict__ kRemA,
                        unsigned* __restrict__ prefixA, int pass) {
  const int b = blockIdx.x;
  __shared__ unsigned h[256];
  h[threadIdx.x] = histCnt[b * 256 + threadIdx.x];
  __syncthreads();
  if (threadIdx.x == 0) {
    unsigned keep, cntAbove, kRem, prefix;
    if (pass == 0) {
      unsigned np = nposA[b], nn = nnegA[b];
      unsigned k3 = 3u * np;
      keep = (nn < k3) ? nn : k3;
      keepA[b] = keep;
      cntAbove = 0u; kRem = keep; prefix = 0u;
    } else {
      keep = keepA[b]; cntAbove = cntAboveA[b];
      kRem = kRemA[b]; prefix = prefixA[b];
    }
    if (keep > 0u) {
      for (int bin = 255; bin >= 0; --bin) {
        unsigned c = h[bin];
        if (c >= kRem) { prefix = (prefix << 8) | (unsigned)bin; break; }
        kRem -= c; cntAbove += c;
      }
    }
    cntAboveA[b] = cntAbove; kRemA[b] = kRem; prefixA[b] = prefix;
  }
  __syncthreads();
  histCnt[b * 256 + threadIdx.x] = 0u;
}

// Sum of neg values strictly above threshold key T; per-block slot partials.
__global__ void ksum(const float* __restrict__ cls,
                     const uint8_t* __restrict__ flags,
                     const unsigned* __restrict__ keepA,
                     const unsigned* __restrict__ prefixA,
                     float* __restrict__ slots) {
  const int b = blockIdx.y;
  const int a = blockIdx.x * kBlk + threadIdx.x;
  __shared__ float sred[8];
  float v = 0.0f;
  if (keepA[b] > 0u) {
    const uint8_t f = flags[(size_t)b * kA + a];
    if (f & 1u) {
      float val = -cls[((size_t)b * kA + a) * 2 + 1];
      if (sortKey(val) > prefixA[b]) v = val;
    }
  }
  float tot = block_reduce256(v, sred);
  if (threadIdx.x == 0) slots[b * kGrid + blockIdx.x] = tot;
}

// bbox smooth-L1 over pos anchors + pos classification mean.
__global__ void kbbox(const float* __restrict__ cls,
                      const float* __restrict__ breg,
                      const float* __restrict__ anchors,
                      const float* __restrict__ ann,
                      const uint8_t* __restrict__ flags,
                      const unsigned* __restrict__ nposA,
                      float* __restrict__ posMeanA,
                      float* __restrict__ out) {
  const int b = blockIdx.x;
  __shared__ float gx1[kN], gy1[kN], gx2[kN], gy2[kN];
  __shared__ float sred[8];
  if (threadIdx.x < kN) {
    const float* p = &ann[((size_t)b * kN + threadIdx.x) * 200];
    gx1[threadIdx.x] = p[0]; gy1[threadIdx.x] = p[1];
    gx2[threadIdx.x] = p[2]; gy2[threadIdx.x] = p[3];
  }
  __syncthreads();
  float accP = 0.0f, accB = 0.0f;
  for (int a = threadIdx.x; a < kA; a += kBlk) {
    const uint8_t f = flags[(size_t)b * kA + a];
    if (f & 2u) {
      const int arg = f >> 2;
      accP += -cls[((size_t)b * kA + a) * 2 + 0];
      const float4 ab = reinterpret_cast<const float4*>(anchors)[a];
      float aw = ab.z - ab.x, ah = ab.w - ab.y;
      float acx = ab.x + 0.5f * aw, acy = ab.y + 0.5f * ah;
      float gw = gx2[arg] - gx1[arg], gh = gy2[arg] - gy1[arg];
      float gcx = gx1[arg] + 0.5f * gw, gcy = gy1[arg] + 0.5f * gh;
      float t0 = (gcx - acx) / (aw + 1e-14f) / 0.1f;
      float t1 = (gcy - acy) / (ah + 1e-14f) / 0.1f;
      float t2 = logf(gw / aw) / 0.2f;
      float t3 = logf(gh / ah) / 0.2f;
      const float* br = &breg[((size_t)b * kA + a) * 4];
      float d, s;
      d = fabsf(t0 - br[0]); s = (d < 1.0f) ? 0.5f * d * d : d - 0.5f; accB += s;
      d = fabsf(t1 - br[1]); s = (d < 1.0f) ? 0.5f * d * d : d - 0.5f; accB += s;
      d = fabsf(t2 - br[2]); s = (d < 1.0f) ? 0.5f * d * d : d - 0.5f; accB += s;
      d = fabsf(t3 - br[3]); s = (d < 1.0f) ? 0.5f * d * d : d - 0.5f; accB += s;
    }
  }
  float sp = block_reduce256(accP, sred);
  float sb = block_reduce256(accB, sred);
  if (threadIdx.x == 0) {
    unsigned np = nposA[b];
    posMeanA[b]  = (np > 0u) ? sp / (float)np : 0.0f;
    out[kB + b]  = (np > 0u) ? sb / ((float)np * 4.0f) : 0.0f;
  }
}

// Landmark wing loss: only gathered pos rows of the 1.28 GB tensor are read.
__global__ void kldm(const float* __restrict__ lreg,
                     const float* __restrict__ anchors,
                     const float* __restrict__ ann,
                     const uint8_t* __restrict__ flags,
                     float* __restrict__ out) {
  const int b = blockIdx.x;
  __shared__ unsigned char hasL[kN];
  __shared__ float sred[8];
  if (threadIdx.x < kN) {
    const float* p = &ann[((size_t)b * kN + threadIdx.x) * 200 + 4];
    float s = 0.0f;
    for (int i = 0; i < kL; ++i) s += p[i];
    hasL[threadIdx.x] = (s > 0.0f) ? 1 : 0;
  }
  __syncthreads();
  float acc = 0.0f, nl = 0.0f;
  for (int a = threadIdx.x; a < kA; a += kBlk) {
    const uint8_t f = flags[(size_t)b * kA + a];
    if (f & 2u) {
      const int arg = f >> 2;
      if (hasL[arg]) {
        nl += 1.0f;
        const float4 ab = reinterpret_cast<const float4*>(anchors)[a];
        float aw = ab.z - ab.x, ah = ab.w - ab.y;
        float acx = ab.x + 0.5f * aw, acy = ab.y + 0.5f * ah;
        float denx = aw + 1e-14f, deny = ah + 1e-14f;
        const float* gl = &ann[((size_t)b * kN + arg) * 200 + 4];
        const float* lr = &lreg[((size_t)b * kA + a) * kL];
        for (int i = 0; i < kL; ++i) {
          bool even = (i & 1) == 0;
          float ctr = even ? acx : acy;
          float den = even ? denx : deny;
          float lt  = (gl[i] - ctr) / den / 0.1f;
          float sc  = (i < 68) ? 1.0f : 3.0f;
          float dd  = fabsf(lt * sc - lr[i] * sc);
          acc += (dd < 3.0f) ? 3.0f * log1pf(dd * 0.5f) : dd - kWingC;
        }
      }
    }
  }
  float sA = block_reduce256(acc, sred);
  float sN = block_reduce256(nl, sred);
  if (threadIdx.x == 0) {
    int n = (int)(sN + 0.5f);
    out[2 * kB + b] = (n > 0) ? sA / ((float)n * (float)kL) : 0.0f;
  }
}

// Combine: neg_mean = (sum_{key>T} + kRem * value(T)) / keep; cls_loss.
__global__ void kfinal(const unsigned* __restrict__ nposA,
                       const unsigned* __restrict__ keepA,
                       const unsigned* __restrict__ kRemA,
                       const unsigned* __restrict__ prefixA,
                       const float* __restrict__ slots,
                       const float* __restrict__ posMeanA,
                       float* __restrict__ out) {
  const int b = blockIdx.x;
  __shared__ float sred[8];
  float v = 0.0f;
  for (int i = threadIdx.x; i < kGrid; i += kBlk) v += slots[b * kGrid + i];
  float sGT = block_reduce256(v, sred);
  if (threadIdx.x == 0) {
    unsigned keep = keepA[b], np = nposA[b];
    float negMean = 0.0f;
    if (keep > 0u) {
      float tval  = invKey(prefixA[b]);
      float total = sGT + (float)kRemA[b] * tval;
      negMean = total / (float)keep;
    }
    out[b] = (np > 0u) ? (posMeanA[b] + negMean) : 0.0f;
  }
}

extern "C" void kernel_launch(void* const* d_in, const int* in_sizes, int n_in,
                              void* d_out, int out_size, void* d_ws,
                              size_t ws_size, hipStream_t stream) {
  (void)in_sizes; (void)n_in; (void)out_size; (void)ws_size;
  const float* cls     = (const float*)d_in[0];
  const float* breg    = (const float*)d_in[1];
  const float* lreg    = (const float*)d_in[2];
  const float* anchors = (const float*)d_in[3];  // (1, A, 4)
  const float* ann     = (const float*)d_in[4];  // (B, 32, 200)
  float* out = (float*)d_out;

  // Scratch carve (~1.7 MB total)
  uint8_t* ws = (uint8_t*)d_ws;
  size_t off = 0;
  uint8_t*  flags   = ws;                          off += (size_t)kB * kA;
  unsigned* histCnt = (unsigned*)(ws + off);       off += (size_t)kB * 256 * 4;
  unsigned* nposA   = (unsigned*)(ws + off);       off += kB * 4;
  unsigned* nnegA   = (unsigned*)(ws + off);       off += kB * 4;
  unsigned* keepA   = (unsigned*)(ws + off);       off += kB * 4;
  unsigned* cntAbA  = (unsigned*)(ws + off);       off += kB * 4;
  unsigned* kRemA   = (unsigned*)(ws + off);       off += kB * 4;
  unsigned* prefixA = (unsigned*)(ws + off);       off += kB * 4;
  float*    posMean = (float*)(ws + off);          off += kB * 4;
  float*    slots   = (float*)(ws + off);          off += (size_t)kB * kGrid * 4;

  // Zero counters + histograms (hist..prefix contiguous: 4096 + 6*16 uints)
  kinit<<<4, 256, 0, stream>>>(histCnt, kB * 256 + 6 * kB);

  dim3 gA(kGrid, kB);
  kiou<<<gA, kBlk, 0, stream>>>(anchors, ann, flags, nposA, nnegA);

  for (int p = 0; p < 4; ++p) {
    khist<<<gA, kBlk, 0, stream>>>(cls, flags, histCnt, keepA, prefixA, p);
    kselect<<<kB, 256, 0, stream>>>(histCnt, nposA, nnegA, keepA, cntAbA,
                                    kRemA, prefixA, p);
  }

  ksum<<<gA, kBlk, 0, stream>>>(cls, flags, keepA, prefixA, slots);
  kbbox<<<kB, kBlk, 0, stream>>>(cls, breg, anchors, ann, flags, nposA,
                                 posMean, out);
  kldm<<<kB, kBlk, 0, stream>>>(lreg, anchors, ann, flags, out);
  kfinal<<<kB, kBlk, 0, stream>>>(nposA, keepA, kRemA, prefixA, slots,
                                  posMean, out);
}